// GAT_54520314855454
// MI455X (gfx1250) — compile-verified
//
#include <hip/hip_runtime.h>
#include <hip/hip_bf16.h>

// ---------------------------------------------------------------------------
// Problem constants (match reference)
// ---------------------------------------------------------------------------
#define GN 50000
#define GE 800000
#define F_IN 512
#define HEADS 8
#define FPH 16
#define C1 (HEADS * FPH)   // 128
#define NCLS 32
#define NEG_SLOPE 0.2f

typedef float v2f __attribute__((ext_vector_type(2)));
typedef float v8f __attribute__((ext_vector_type(8)));

// ---------------------------------------------------------------------------
// monotone float <-> uint encoding so atomicMax(u32) == float max
// encode(-huge) is small positive; 0u is below every real encoding (== -inf id)
// ---------------------------------------------------------------------------
__device__ __forceinline__ unsigned enc_f32(float f) {
  unsigned u = __float_as_uint(f);
  return (u & 0x80000000u) ? ~u : (u | 0x80000000u);
}
__device__ __forceinline__ float dec_f32(unsigned u) {
  return (u & 0x80000000u) ? __uint_as_float(u ^ 0x80000000u) : __uint_as_float(~u);
}
__device__ __forceinline__ float leaky(float v) { return v > 0.f ? v : NEG_SLOPE * v; }

// ---------------------------------------------------------------------------
// CDNA5 async global->LDS copy (ASYNCcnt-tracked, no VGPR data path).
// lds_byte: wave-relative LDS byte address (low 32 bits of generic pointer)
// ---------------------------------------------------------------------------
__device__ __forceinline__ void async_copy_b128(unsigned lds_byte, const void* gaddr) {
  asm volatile("global_load_async_to_lds_b128 %0, %1, off"
               :: "v"(lds_byte), "v"(gaddr) : "memory");
}
__device__ __forceinline__ void wait_async0() {
#if __has_builtin(__builtin_amdgcn_s_wait_asynccnt)
  __builtin_amdgcn_s_wait_asynccnt(0);
#else
  asm volatile("s_wait_asynccnt 0x0" ::: "memory");
#endif
}

// ---------------------------------------------------------------------------
// Repack W[K][NCOL] -> Wp[K/2][NCOL][2] so a B fragment {W[k][c], W[k+1][c]}
// is one aligned 8-byte LDS load straight into an even VGPR pair.
// ---------------------------------------------------------------------------
template <int K, int NCOL>
__global__ void gat_pack_pairs(const float* __restrict__ W, float* __restrict__ Wp) {
  int t = blockIdx.x * blockDim.x + threadIdx.x;
  if (t >= (K / 2) * NCOL) return;
  int p = t / NCOL, col = t - p * NCOL;
  float2 v;
  v.x = W[(2 * p) * NCOL + col];
  v.y = W[(2 * p + 1) * NCOL + col];
  ((float2*)Wp)[t] = v;
}

// ---------------------------------------------------------------------------
// WMMA GEMM: C[nrows, NCOL] = A[nrows, K] * Bp[K, NCOL]   (f32, 16x16x4 WMMA)
// Bp is the pair-interleaved weight. 256 threads = 8 waves; each wave owns a
// 16-row stripe and NCOL/16 col tiles. Bp staged through LDS in KC-row chunks
// via async-to-LDS DMA (chunk of Bp is a contiguous byte range).
// ---------------------------------------------------------------------------
template <int K, int NCOL>
__global__ __launch_bounds__(256) void gat_gemm_wmma(const float* __restrict__ A,
                                                     const float* __restrict__ Bp,
                                                     float* __restrict__ C, int nrows) {
  constexpr int KC = 64;
  constexpr int NT = NCOL / 16;
  __shared__ v2f lB[(KC / 2) * NCOL];               // pair-interleaved chunk
  const unsigned ldsB = (unsigned)(size_t)(&lB[0]); // LDS byte offset

  const int wave = threadIdx.x >> 5;
  const int lane = threadIdx.x & 31;
  const int half = lane >> 4;          // 0: lanes 0-15, 1: lanes 16-31
  const int l16  = lane & 15;
  const int rowBase = blockIdx.x * 128 + wave * 16;
  const int mrow = rowBase + l16;
  // OOB rows read row 0; their products land only in OOB output rows,
  // which are never stored, so no zeroing is needed.
  const float* __restrict__ arow = A + (size_t)(mrow < nrows ? mrow : 0) * K;
  const int koff = half * 2;           // this half-wave supplies K= koff, koff+1

  v8f acc[NT];
#pragma unroll
  for (int t = 0; t < NT; ++t) acc[t] = v8f{0.f, 0.f, 0.f, 0.f, 0.f, 0.f, 0.f, 0.f};

  for (int k0 = 0; k0 < K; k0 += KC) {
    __syncthreads();   // WAR: previous chunk fully consumed before DMA overwrites
    const float4* s4 = (const float4*)(Bp + (size_t)(k0 / 2) * NCOL * 2);
#pragma unroll 2
    for (int i = threadIdx.x; i < KC * NCOL / 4; i += 256)
      async_copy_b128(ldsB + (unsigned)i * 16u, (const void*)(s4 + i));
    wait_async0();
    __syncthreads();

#pragma unroll 4
    for (int kk = 0; kk < KC; kk += 4) {
      // A fragment: lane(l16) = row M, half selects K pair {0,1} or {2,3}
      const float2 av = *(const float2*)(arow + k0 + kk + koff);
      v2f a;
      a.x = av.x;
      a.y = av.y;
      const int prow = ((kk + koff) >> 1) * NCOL;   // pair-row base in LDS
#pragma unroll
      for (int t = 0; t < NT; ++t) {
        const v2f b = lB[prow + t * 16 + l16];      // one ds_load_b64
        acc[t] = __builtin_amdgcn_wmma_f32_16x16x4_f32(false, a, false, b,
                                                       (short)0, acc[t], false, false);
      }
    }
  }

  // D layout: VGPR v -> M = v (lanes 0-15) / v+8 (lanes 16-31), N = lane&15
#pragma unroll
  for (int t = 0; t < NT; ++t) {
#pragma unroll
    for (int v = 0; v < 8; ++v) {
      const int row = rowBase + v + (half << 3);
      if (row < nrows) C[(size_t)row * NCOL + t * 16 + l16] = acc[t][v];
    }
  }
}

// ---------------------------------------------------------------------------
// a_src[n,h] = sum_c h[n,h,c]*att_src[h,c]   (and same for att_dst)
// ---------------------------------------------------------------------------
__global__ void gat_att_scores(const float* __restrict__ h,
                               const float* __restrict__ att_src,
                               const float* __restrict__ att_dst,
                               float* __restrict__ a_src, float* __restrict__ a_dst,
                               int n, int H, int C) {
  int t = blockIdx.x * blockDim.x + threadIdx.x;
  if (t >= n * H) return;
  int hh = t % H;
  const float* hp = h + (size_t)t * C;           // t == node*H + hh, layout [N,H,C]
  const float* as = att_src + hh * C;
  const float* ad = att_dst + hh * C;
  float s = 0.f, d = 0.f;
  for (int c = 0; c < C; ++c) { float v = hp[c]; s += v * as[c]; d += v * ad[c]; }
  a_src[t] = s;
  a_dst[t] = d;
}

// ---------------------------------------------------------------------------
// edge pass 1: segment max of leaky(a_src[src]+a_dst[dst]) into encoded u32
// ---------------------------------------------------------------------------
__global__ void gat_edge_max(const int* __restrict__ src, const int* __restrict__ dst,
                             const float* __restrict__ a_src, const float* __restrict__ a_dst,
                             unsigned* __restrict__ emax, int E, int H) {
  int t = blockIdx.x * blockDim.x + threadIdx.x;
  if (t >= E * H) return;
  int e = t / H, hh = t - e * H;
  float v = leaky(a_src[src[e] * H + hh] + a_dst[dst[e] * H + hh]);
  atomicMax(&emax[dst[e] * H + hh], enc_f32(v));
}

// ---------------------------------------------------------------------------
// edge pass 2: ex = exp(e - emax[dst]); store ex, atomicAdd denom[dst]
// ---------------------------------------------------------------------------
__global__ void gat_edge_expsum(const int* __restrict__ src, const int* __restrict__ dst,
                                const float* __restrict__ a_src, const float* __restrict__ a_dst,
                                const unsigned* __restrict__ emax, float* __restrict__ denom,
                                float* __restrict__ exbuf, int E, int H) {
  int t = blockIdx.x * blockDim.x + threadIdx.x;
  if (t >= E * H) return;
  int e = t / H, hh = t - e * H;
  int d = dst[e];
  float v = leaky(a_src[src[e] * H + hh] + a_dst[d * H + hh]);
  float ex = __expf(v - dec_f32(emax[d * H + hh]));
  exbuf[t] = ex;
  atomicAdd(&denom[d * H + hh], ex);
}

// ---------------------------------------------------------------------------
// edge pass 3: agg[dst,h,c] += h[src,h,c] * (ex/denom[dst,h]); 1 thread/(e,h,c)
// ---------------------------------------------------------------------------
__global__ void gat_edge_aggregate(const int* __restrict__ src, const int* __restrict__ dst,
                                   const float* __restrict__ exbuf,
                                   const float* __restrict__ denom,
                                   const float* __restrict__ h, float* __restrict__ agg,
                                   int E, int H, int C) {
  int t = blockIdx.x * blockDim.x + threadIdx.x;
  if (t >= E * H * C) return;
  int c = t % C;
  int eh = t / C;
  int e = eh / H, hh = eh - e * H;
  int s = src[e], d = dst[e];
  float alpha = exbuf[eh] / fmaxf(denom[d * H + hh], 1e-16f);
  atomicAdd(&agg[((size_t)d * H + hh) * C + c],
            h[((size_t)s * H + hh) * C + c] * alpha);
}

// ---------------------------------------------------------------------------
// in-place x = elu(x + bias[j])
// ---------------------------------------------------------------------------
__global__ void gat_bias_elu(float* __restrict__ a, const float* __restrict__ b,
                             int n, int F) {
  int t = blockIdx.x * blockDim.x + threadIdx.x;
  if (t >= n * F) return;
  float v = a[t] + b[t % F];
  a[t] = v > 0.f ? v : (__expf(v) - 1.f);
}

// ---------------------------------------------------------------------------
// out[n,:] = log_softmax(agg2[n,:] + b2)
// ---------------------------------------------------------------------------
__global__ void gat_bias_logsoftmax(const float* __restrict__ agg2,
                                    const float* __restrict__ b2,
                                    float* __restrict__ out, int n) {
  int node = blockIdx.x * blockDim.x + threadIdx.x;
  if (node >= n) return;
  const float* p = agg2 + (size_t)node * NCLS;
  float v[NCLS];
  float mx = -3.4e38f;
#pragma unroll
  for (int j = 0; j < NCLS; ++j) { v[j] = p[j] + b2[j]; mx = fmaxf(mx, v[j]); }
  float s = 0.f;
#pragma unroll
  for (int j = 0; j < NCLS; ++j) s += __expf(v[j] - mx);
  float lse = mx + __logf(s);
  float* o = out + (size_t)node * NCLS;
#pragma unroll
  for (int j = 0; j < NCLS; ++j) o[j] = v[j] - lse;
}

// ---------------------------------------------------------------------------
// launch
// ---------------------------------------------------------------------------
static inline int cdiv(long long a, int b) { return (int)((a + b - 1) / b); }

extern "C" void kernel_launch(void* const* d_in, const int* in_sizes, int n_in,
                              void* d_out, int out_size, void* d_ws, size_t ws_size,
                              hipStream_t stream) {
  const float* x        = (const float*)d_in[0];
  const float* W1       = (const float*)d_in[1];
  const float* att_src1 = (const float*)d_in[2];
  const float* att_dst1 = (const float*)d_in[3];
  const float* b1       = (const float*)d_in[4];
  const float* W2       = (const float*)d_in[5];
  const float* att_src2 = (const float*)d_in[6];
  const float* att_dst2 = (const float*)d_in[7];
  const float* b2       = (const float*)d_in[8];
  const int*   eidx     = (const int*)d_in[9];
  const int* src = eidx;
  const int* dst = eidx + GE;
  float* out = (float*)d_out;

  // workspace layout (floats / u32, all 4-byte)
  float* w = (float*)d_ws;
  size_t o = 0;
  float*    h1     = w + o; o += (size_t)GN * C1;     // 6.4M
  float*    agg1   = w + o; o += (size_t)GN * C1;     // 6.4M
  float*    asrc1  = w + o; o += (size_t)GN * HEADS;
  float*    adst1  = w + o; o += (size_t)GN * HEADS;
  unsigned* emax1  = (unsigned*)(w + o); o += (size_t)GN * HEADS;
  float*    den1   = w + o; o += (size_t)GN * HEADS;
  float*    ex1    = w + o; o += (size_t)GE * HEADS;  // 6.4M
  float*    h2     = w + o; o += (size_t)GN * NCLS;
  float*    agg2   = w + o; o += (size_t)GN * NCLS;
  float*    asrc2  = w + o; o += (size_t)GN;
  float*    adst2  = w + o; o += (size_t)GN;
  unsigned* emax2  = (unsigned*)(w + o); o += (size_t)GN;
  float*    den2   = w + o; o += (size_t)GN;
  float*    ex2    = w + o; o += (size_t)GE;
  float*    Wp1    = w + o; o += (size_t)F_IN * C1;   // pair-interleaved W1
  float*    Wp2    = w + o; o += (size_t)C1 * NCLS;   // pair-interleaved W2

  const int B = 256;

  // weight repack (tiny)
  gat_pack_pairs<F_IN, C1><<<cdiv((F_IN / 2) * C1, B), B, 0, stream>>>(W1, Wp1);
  gat_pack_pairs<C1, NCLS><<<cdiv((C1 / 2) * NCLS, B), B, 0, stream>>>(W2, Wp2);

  // zero-init accumulators (encoded -inf == 0u, so memset 0 works for all)
  hipMemsetAsync(emax1, 0, (size_t)GN * HEADS * 4, stream);
  hipMemsetAsync(den1,  0, (size_t)GN * HEADS * 4, stream);
  hipMemsetAsync(agg1,  0, (size_t)GN * C1 * 4, stream);
  hipMemsetAsync(emax2, 0, (size_t)GN * 4, stream);
  hipMemsetAsync(den2,  0, (size_t)GN * 4, stream);
  hipMemsetAsync(agg2,  0, (size_t)GN * NCLS * 4, stream);

  // ----- layer 1 -----
  gat_gemm_wmma<F_IN, C1><<<cdiv(GN, 128), B, 0, stream>>>(x, Wp1, h1, GN);
  gat_att_scores<<<cdiv((long long)GN * HEADS, B), B, 0, stream>>>(
      h1, att_src1, att_dst1, asrc1, adst1, GN, HEADS, FPH);
  gat_edge_max<<<cdiv((long long)GE * HEADS, B), B, 0, stream>>>(
      src, dst, asrc1, adst1, emax1, GE, HEADS);
  gat_edge_expsum<<<cdiv((long long)GE * HEADS, B), B, 0, stream>>>(
      src, dst, asrc1, adst1, emax1, den1, ex1, GE, HEADS);
  gat_edge_aggregate<<<cdiv((long long)GE * HEADS * FPH, B), B, 0, stream>>>(
      src, dst, ex1, den1, h1, agg1, GE, HEADS, FPH);
  gat_bias_elu<<<cdiv((long long)GN * C1, B), B, 0, stream>>>(agg1, b1, GN, C1);

  // ----- layer 2 -----
  gat_gemm_wmma<C1, NCLS><<<cdiv(GN, 128), B, 0, stream>>>(agg1, Wp2, h2, GN);
  gat_att_scores<<<cdiv(GN, B), B, 0, stream>>>(
      h2, att_src2, att_dst2, asrc2, adst2, GN, 1, NCLS);
  gat_edge_max<<<cdiv(GE, B), B, 0, stream>>>(src, dst, asrc2, adst2, emax2, GE, 1);
  gat_edge_expsum<<<cdiv(GE, B), B, 0, stream>>>(src, dst, asrc2, adst2, emax2, den2, ex2, GE, 1);
  gat_edge_aggregate<<<cdiv((long long)GE * NCLS, B), B, 0, stream>>>(
      src, dst, ex2, den2, h2, agg2, GE, 1, NCLS);

  gat_bias_logsoftmax<<<cdiv(GN, B), B, 0, stream>>>(agg2, b2, out, GN);
}